// dynamic_model_52390011077268
// MI455X (gfx1250) — compile-verified
//
#include <hip/hip_runtime.h>
#include <math.h>
#include <stdint.h>

// Problem constants (from reference): B=8, N=1024.
#define NMAT 1024
#define BB 8
#define NS_ITERS 30

typedef float v2f __attribute__((ext_vector_type(2)));
typedef float v8f __attribute__((ext_vector_type(8)));
typedef unsigned int v4u __attribute__((ext_vector_type(4)));
typedef int v8i __attribute__((ext_vector_type(8)));
typedef int v4i __attribute__((ext_vector_type(4)));

// ---------------------------------------------------------------------------
// trace(M) where M = N*|P_u|  -> single float in *s
// ---------------------------------------------------------------------------
__global__ __launch_bounds__(256) void trace_kernel(const float* __restrict__ Pu,
                                                    float* __restrict__ s) {
    __shared__ float red[256];
    int t = threadIdx.x;
    float v = 0.0f;
    for (int i = t; i < NMAT; i += 256)
        v += (float)NMAT * fabsf(Pu[i * NMAT + i]);
    red[t] = v;
    __syncthreads();
    for (int o = 128; o > 0; o >>= 1) {
        if (t < o) red[t] += red[t + o];
        __syncthreads();
    }
    if (t == 0) *s = red[0];
}

// ---------------------------------------------------------------------------
// Y0 = N*|P_u| / trace ;  Z0 = I
// ---------------------------------------------------------------------------
__global__ __launch_bounds__(256) void init_yz_kernel(const float* __restrict__ Pu,
                                                      const float* __restrict__ s,
                                                      float* __restrict__ Y,
                                                      float* __restrict__ Z) {
    int idx = blockIdx.x * 256 + threadIdx.x;  // over N*N
    float inv = 1.0f / (*s);
    Y[idx] = (float)NMAT * fabsf(Pu[idx]) * inv;
    int r = idx >> 10;
    int c = idx & (NMAT - 1);
    Z[idx] = (r == c) ? 1.0f : 0.0f;
}

// ---------------------------------------------------------------------------
// TDM 2-D tile load descriptor (D#) per CDNA5 ISA.
// Copies tile_d1 rows x tile_d0 elements (fp32) from global (row stride
// stride0 elements) into LDS at lds_addr; optional LDS row padding of 1 dword
// every 16 dwords (pad=true) to avoid bank conflicts.
// ---------------------------------------------------------------------------
__device__ __forceinline__ void tdm_load_2d(unsigned lds_addr, const float* gptr,
                                            unsigned tile_d0, unsigned tile_d1,
                                            unsigned tensor_d0, unsigned tensor_d1,
                                            unsigned stride0, bool pad) {
    unsigned long long ga = (unsigned long long)(uintptr_t)gptr;
    v4u g0;
    g0[0] = 1u;                                      // count=1 (valid), user mode
    g0[1] = lds_addr;                                // lds_addr[31:0]
    g0[2] = (unsigned)(ga & 0xFFFFFFFFu);            // global_addr[31:0]
    g0[3] = (unsigned)((ga >> 32) & 0x01FFFFFFu)     // global_addr[56:32]
          | (2u << 30);                              // type=2 ("image")
    unsigned dw0 = (2u << 16);                       // data_size=2 -> 4 bytes
    if (pad) dw0 |= (1u << 20)                       // pad_enable
               | (3u << 22)                          // pad_interval: 16 dwords
               | (0u << 25);                         // pad_amount: 1 dword
    v8i g1;
    g1[0] = (int)dw0;                                // wg_mask=0, flags
    g1[1] = (int)((tensor_d0 & 0xFFFFu) << 16);      // abar=0 | td0[15:0]
    g1[2] = (int)(((tensor_d0 >> 16) & 0xFFFFu) | ((tensor_d1 & 0xFFFFu) << 16));
    g1[3] = (int)(((tensor_d1 >> 16) & 0xFFFFu) | ((tile_d0 & 0xFFFFu) << 16));
    g1[4] = (int)(tile_d1 & 0xFFFFu);                // tile_dim1 | tile_dim2=0
    g1[5] = (int)stride0;                            // tensor_dim0_stride[31:0]
    g1[6] = 0;                                       // stride hi, dim1_stride lo
    g1[7] = 0;
    v4i g2 = {0, 0, 0, 0};
    v4i g3 = {0, 0, 0, 0};
    v8i g4 = {0, 0, 0, 0, 0, 0, 0, 0};
    // amdgpu-toolchain (clang-23) 6-arg form:
    // (uint32x4, int32x8, int32x4, int32x4, int32x8, i32 cpol)
    __builtin_amdgcn_tensor_load_to_lds(g0, g1, g2, g3, g4, 0);
}

// ---------------------------------------------------------------------------
// FP32 WMMA GEMM with TDM->LDS double-buffered staging:
//   D = alpha * (A @ B) + beta * C      (C may be nullptr)
// Block: 128 threads (4 waves), 64x64 block tile, 32x32 per wave.
// K stepped 16 per stage; wave 0 drives the Tensor Data Mover.
// ---------------------------------------------------------------------------
#define KB 16
#define AROW 17  // 16 dwords + 1 pad dword (TDM pad) -> conflict-free LDS reads

__global__ __launch_bounds__(128) void gemm_ns_kernel(float* __restrict__ D,
                                                      const float* __restrict__ A,
                                                      const float* __restrict__ B,
                                                      const float* __restrict__ C,
                                                      float alpha, float beta) {
    __shared__ float shA[2][64 * AROW];  // 64 rows x (16+1) floats
    __shared__ float shB[2][KB * 64];    // 16 rows x 64 floats

    const int lane  = threadIdx.x & 31;
    const int wave  = threadIdx.x >> 5;
    const int l15   = lane & 15;
    const int khalf = (lane >> 4) << 1;      // lanes 0-15 -> K 0,1 ; 16-31 -> K 2,3
    const int brow0 = blockIdx.y * 64;       // block row origin
    const int bcol0 = blockIdx.x * 64;       // block col origin
    const int rloc  = (wave >> 1) * 32;      // wave row offset in block tile
    const int cloc  = (wave & 1) * 32;       // wave col offset in block tile

    const unsigned ldsA0 = (unsigned)(uintptr_t)(void*)&shA[0][0];
    const unsigned ldsA1 = (unsigned)(uintptr_t)(void*)&shA[1][0];
    const unsigned ldsB0 = (unsigned)(uintptr_t)(void*)&shB[0][0];
    const unsigned ldsB1 = (unsigned)(uintptr_t)(void*)&shB[1][0];

    v8f acc00 = {}; v8f acc01 = {}; v8f acc10 = {}; v8f acc11 = {};

    // Prologue: stage k=0 into buffer 0 via TDM, wave 0 only.
    if (wave == 0) {
        // A tile: 64 rows x 16 cols, row stride NMAT, LDS padded rows.
        tdm_load_2d(ldsA0, A + (size_t)brow0 * NMAT, KB, 64, NMAT, NMAT, NMAT, true);
        // B tile: 16 rows x 64 cols.
        tdm_load_2d(ldsB0, B + bcol0, 64, KB, NMAT, NMAT, NMAT, false);
        __builtin_amdgcn_s_wait_tensorcnt(0);
    }
    __syncthreads();

    for (int kb = 0; kb < NMAT; kb += KB) {
        const int cur = (kb >> 4) & 1;
        const bool more = (kb + KB) < NMAT;
        if (more && wave == 0) {
            const unsigned la = cur ? ldsA1 : ldsA0;
            const unsigned lb = cur ? ldsB1 : ldsB0;
            // note: 'cur' buffer is being computed from; stage into the other
            tdm_load_2d(cur ? ldsA0 : ldsA1, A + (size_t)brow0 * NMAT + (kb + KB), KB, 64, NMAT, NMAT, NMAT, true);
            tdm_load_2d(cur ? ldsB0 : ldsB1, B + (size_t)(kb + KB) * NMAT + bcol0, 64, KB, NMAT, NMAT, NMAT, false);
            (void)la; (void)lb;
        }

        const float* __restrict__ sA = &shA[cur][0];
        const float* __restrict__ sB = &shB[cur][0];
#pragma unroll
        for (int kk = 0; kk < KB; kk += 4) {
            const int kf = kk + khalf;
            v2f a0, a1, b0, b1;
            const float* ap0 = sA + (rloc + l15) * AROW + kf;
            const float* ap1 = ap0 + 16 * AROW;
            a0.x = ap0[0]; a0.y = ap0[1];
            a1.x = ap1[0]; a1.y = ap1[1];
            const float* bp = sB + kf * 64 + cloc + l15;
            b0.x = bp[0];  b0.y = bp[64];
            b1.x = bp[16]; b1.y = bp[64 + 16];
            acc00 = __builtin_amdgcn_wmma_f32_16x16x4_f32(false, a0, false, b0, (short)0, acc00, false, false);
            acc01 = __builtin_amdgcn_wmma_f32_16x16x4_f32(false, a0, false, b1, (short)0, acc01, false, false);
            acc10 = __builtin_amdgcn_wmma_f32_16x16x4_f32(false, a1, false, b0, (short)0, acc10, false, false);
            acc11 = __builtin_amdgcn_wmma_f32_16x16x4_f32(false, a1, false, b1, (short)0, acc11, false, false);
        }

        if (more && wave == 0) __builtin_amdgcn_s_wait_tensorcnt(0);
        __syncthreads();  // publish next buffer; protect current from overwrite
    }

    // C/D layout: VGPR v holds row v (lanes 0-15) / row v+8 (lanes 16-31), col=l15
    const int rofs = (lane >> 4) << 3;  // 0 or 8
    const int row0 = brow0 + rloc;
    const int col0 = bcol0 + cloc;
#pragma unroll
    for (int v = 0; v < 8; ++v) {
        int r0 = row0 + rofs + v;
        int r1 = r0 + 16;
        int c0 = col0 + l15;
        int c1 = c0 + 16;
        float d00 = alpha * acc00[v];
        float d01 = alpha * acc01[v];
        float d10 = alpha * acc10[v];
        float d11 = alpha * acc11[v];
        if (C != nullptr) {
            d00 += beta * C[(size_t)r0 * NMAT + c0];
            d01 += beta * C[(size_t)r0 * NMAT + c1];
            d10 += beta * C[(size_t)r1 * NMAT + c0];
            d11 += beta * C[(size_t)r1 * NMAT + c1];
        }
        D[(size_t)r0 * NMAT + c0] = d00;
        D[(size_t)r0 * NMAT + c1] = d01;
        D[(size_t)r1 * NMAT + c0] = d10;
        D[(size_t)r1 * NMAT + c1] = d11;
    }
}

// ---------------------------------------------------------------------------
// P_k[b,n,m] = d_b[n]*d_b[m] + |P_u[n,m]| + 0.01*Q_k[n,m]^2
//   with d_b[n] = x[0,0,n,0] - x[0,b,n,0]   (x flat: [B,N])
// ---------------------------------------------------------------------------
__global__ __launch_bounds__(256) void pk_kernel(const float* __restrict__ x,
                                                 const float* __restrict__ Pu,
                                                 const float* __restrict__ Qk,
                                                 float* __restrict__ out) {
    int idx = blockIdx.x * 256 + threadIdx.x;  // over B*N*N = 8M
    int m = idx & (NMAT - 1);
    int n = (idx >> 10) & (NMAT - 1);
    int b = idx >> 20;
    float dn = x[n] - x[b * NMAT + n];
    float dm = x[m] - x[b * NMAT + m];
    float q  = Qk[n * NMAT + m];
    out[idx] = dn * dm + fabsf(Pu[n * NMAT + m]) + 0.01f * q * q;
}

// ---------------------------------------------------------------------------
// sigmaT[0,b,j,n] = x[b,n] + (j<N ? +p[n,j] : -p[n,j-N]),  p = sqrt(s)*Y
// p is symmetric (polynomial of symmetric Y0), so read Y[jm*N + n] (coalesced).
// ---------------------------------------------------------------------------
__global__ __launch_bounds__(256) void sigma_kernel(const float* __restrict__ x,
                                                    const float* __restrict__ Y,
                                                    const float* __restrict__ s,
                                                    float* __restrict__ out2) {
    unsigned idx = blockIdx.x * 256 + threadIdx.x;  // over B*2N*N = 16M
    int n  = idx & (NMAT - 1);
    int j  = (idx >> 10) & (2 * NMAT - 1);
    int b  = idx >> 21;
    int jm = j & (NMAT - 1);
    float sq   = sqrtf(*s);
    float sign = (j < NMAT) ? 1.0f : -1.0f;
    float pv   = sq * Y[(size_t)jm * NMAT + n];
    out2[idx]  = x[b * NMAT + n] + sign * pv;
}

// ---------------------------------------------------------------------------
extern "C" void kernel_launch(void* const* d_in, const int* in_sizes, int n_in,
                              void* d_out, int out_size, void* d_ws, size_t ws_size,
                              hipStream_t stream) {
    const float* x  = (const float*)d_in[0];   // [1,B,N,1] -> flat [B,N]
    const float* Pu = (const float*)d_in[1];   // [N,N]
    const float* Qk = (const float*)d_in[2];   // [N,N]

    float* out    = (float*)d_out;
    float* outPk  = out;                                  // B*N*N
    float* outSig = out + (size_t)BB * NMAT * NMAT;       // B*2N*N

    const size_t NN = (size_t)NMAT * NMAT;
    float* s    = (float*)d_ws;
    float* base = (float*)((char*)d_ws + 256);
    float* pY = base;
    float* pZ = base + NN;
    float* pT = base + 2 * NN;
    float* pU = base + 3 * NN;

    trace_kernel<<<1, 256, 0, stream>>>(Pu, s);
    init_yz_kernel<<<(unsigned)(NN / 256), 256, 0, stream>>>(Pu, s, pY, pZ);

    dim3 gblk(128);
    dim3 ggrd(NMAT / 64, NMAT / 64);
    for (int it = 0; it < NS_ITERS; ++it) {
        // T = Z @ Y
        gemm_ns_kernel<<<ggrd, gblk, 0, stream>>>(pT, pZ, pY, nullptr, 1.0f, 0.0f);
        // Ynew = 1.5*Y - 0.5*(Y @ T)   -> pU
        gemm_ns_kernel<<<ggrd, gblk, 0, stream>>>(pU, pY, pT, pY, -0.5f, 1.5f);
        // Znew = 1.5*Z - 0.5*(T @ Z)   -> old Y buffer (not an input here)
        gemm_ns_kernel<<<ggrd, gblk, 0, stream>>>(pY, pT, pZ, pZ, -0.5f, 1.5f);
        // rotate: Y <- pU, Z <- (old Y buf), scratch <- old Z / old T
        float* nY = pU; float* nZ = pY; float* nT = pZ; float* nU = pT;
        pY = nY; pZ = nZ; pT = nT; pU = nU;
    }

    pk_kernel<<<(unsigned)((BB * NN) / 256), 256, 0, stream>>>(x, Pu, Qk, outPk);
    sigma_kernel<<<(unsigned)((BB * 2 * NN) / 256), 256, 0, stream>>>(x, pY, s, outSig);
}